// PointInstanceNorm_85023172591852
// MI455X (gfx1250) — compile-verified
//
#include <hip/hip_runtime.h>
#include <hip/hip_bf16.h>

typedef __attribute__((ext_vector_type(2))) float v2f;
typedef __attribute__((ext_vector_type(4))) float v4f;
typedef __attribute__((ext_vector_type(8))) float v8f;

#define DIMS    128
#define PIN_EPS 1e-5f
#define CHUNK   1024   // points per block in pass 1
#define T1      256    // 8 waves; wave w owns dims [16w, 16w+16)

// ---------------------------------------------------------------------------
// Pass 1: per-(segment, chunk) partial sum / sum-of-squares via WMMA.
// A = ones(16x4)  =>  D[m][n] = sum_k B[k][n]  (column sums, accumulated in C).
// lane L: n = L&15 (column / dim), h = L>>4 selects the point-pair it feeds.
// A-is-ones makes the result invariant to the B K-permutation, so only the
// N<->lane mapping matters.
// ---------------------------------------------------------------------------
__global__ void pin_reduce(const float* __restrict__ x,
                           const int*   __restrict__ seqlen,
                           float* __restrict__ gsum,   // [B][DIMS]
                           float* __restrict__ gsqs,   // [B][DIMS]
                           int blocksPerSeg)
{
    const int seg   = blockIdx.x / blocksPerSeg;
    const int chunk = blockIdx.x % blocksPerSeg;
    const int segStart = seqlen[seg];
    const int segEnd   = seqlen[seg + 1];
    const int p0 = segStart + chunk * CHUNK;
    if (p0 >= segEnd) return;
    const int npts = min(CHUNK, segEnd - p0);

    const int tid     = threadIdx.x;
    const int wave    = tid >> 5;        // 0..7
    const int lane    = tid & 31;
    const int n       = lane & 15;       // column within 16-dim group
    const int h       = lane >> 4;       // which point-pair this lane feeds
    const int dimbase = wave * 16;

    const float* ptr = x + (size_t)(p0 + 2 * h) * DIMS + dimbase + n;

    const v2f a = {1.0f, 1.0f};          // all-ones A matrix (16x4 f32)
    v8f csum = {};
    v8f csqs = {};

    const int full = npts >> 2;          // full 4-point steps
    const int rem  = npts & 3;

    for (int t = 0; t < full; ++t) {
        float b0 = __builtin_nontemporal_load(ptr);
        float b1 = __builtin_nontemporal_load(ptr + DIMS);
        ptr += 4 * DIMS;
        v2f b;  b.x  = b0;      b.y  = b1;
        v2f bq; bq.x = b0 * b0; bq.y = b1 * b1;
        csum = __builtin_amdgcn_wmma_f32_16x16x4_f32(false, a, false, b,
                                                     (short)0, csum, false, false);
        csqs = __builtin_amdgcn_wmma_f32_16x16x4_f32(false, a, false, bq,
                                                     (short)0, csqs, false, false);
    }
    if (rem) {                            // uniform per block; EXEC stays full
        const int r0 = 2 * h;
        float b0 = (r0     < rem) ? ptr[0]    : 0.0f;
        float b1 = (r0 + 1 < rem) ? ptr[DIMS] : 0.0f;
        v2f b;  b.x  = b0;      b.y  = b1;
        v2f bq; bq.x = b0 * b0; bq.y = b1 * b1;
        csum = __builtin_amdgcn_wmma_f32_16x16x4_f32(false, a, false, b,
                                                     (short)0, csum, false, false);
        csqs = __builtin_amdgcn_wmma_f32_16x16x4_f32(false, a, false, bq,
                                                     (short)0, csqs, false, false);
    }

    // All D rows are identical (A ones); lane L's c[0] = column sum for dim
    // dimbase + (L&15), duplicated in lanes 0-15 (M=0) and 16-31 (M=8).
    if (h == 0) {
        atomicAdd(&gsum[seg * DIMS + dimbase + n], csum[0]);
        atomicAdd(&gsqs[seg * DIMS + dimbase + n], csqs[0]);
    }
}

// ---------------------------------------------------------------------------
// Pass 2 (tiny): mean/var -> fused scale/shift:  out = x*scale + shift
// ---------------------------------------------------------------------------
__global__ void pin_finalize(const int*   __restrict__ seqlen,
                             const float* __restrict__ gsum,
                             const float* __restrict__ gsqs,
                             const float* __restrict__ weight,
                             const float* __restrict__ bias,
                             float* __restrict__ scale,  // [B][DIMS]
                             float* __restrict__ shift,  // [B][DIMS]
                             int B)
{
    const int i = blockIdx.x * blockDim.x + threadIdx.x;
    if (i >= B * DIMS) return;
    const int b = i / DIMS;
    const int d = i % DIMS;
    const float cnt   = (float)(seqlen[b + 1] - seqlen[b]);
    const float denom = fmaxf(cnt, 1.0f);              // scatter_mean clamp
    const float m   = gsum[i] / denom;
    float       var = gsqs[i] / denom - m * m;
    var = fmaxf(var, 0.0f);
    const float inv = 1.0f / sqrtf(var + PIN_EPS);
    const float sc  = inv * weight[d];
    scale[i] = sc;
    shift[i] = bias[d] - m * sc;
}

// ---------------------------------------------------------------------------
// Pass 3: pure bandwidth stream, float4, non-temporal in & out.
// ---------------------------------------------------------------------------
__global__ void pin_normalize(const float* __restrict__ x,
                              const int*   __restrict__ seqlen,
                              const float* __restrict__ scale,
                              const float* __restrict__ shift,
                              float* __restrict__ out,
                              int B, int nvec)          // nvec = N*DIMS/4
{
    __shared__ int soff[64];                            // B+1 <= 64
    for (int i = threadIdx.x; i <= B; i += blockDim.x) soff[i] = seqlen[i];
    __syncthreads();

    const int v = blockIdx.x * blockDim.x + threadIdx.x;
    if (v >= nvec) return;

    const int point = v >> 5;           // DIMS/4 = 32 float4 per point
    const int d4    = v & 31;

    int s = 0;
    while (soff[s + 1] <= point) ++s;   // B is tiny; block is seg-coherent

    const v4f xv = __builtin_nontemporal_load((const v4f*)x + v);
    const v4f sc = ((const v4f*)scale)[s * 32 + d4];    // hot in L0
    const v4f sh = ((const v4f*)shift)[s * 32 + d4];
    v4f o;
    o.x = xv.x * sc.x + sh.x;
    o.y = xv.y * sc.y + sh.y;
    o.z = xv.z * sc.z + sh.z;
    o.w = xv.w * sc.w + sh.w;
    __builtin_nontemporal_store(o, (v4f*)out + v);
}

// ---------------------------------------------------------------------------
extern "C" void kernel_launch(void* const* d_in, const int* in_sizes, int n_in,
                              void* d_out, int out_size, void* d_ws, size_t ws_size,
                              hipStream_t stream)
{
    const float* x      = (const float*)d_in[0];
    const int*   seqlen = (const int*)  d_in[1];
    const float* weight = (const float*)d_in[2];
    const float* bias   = (const float*)d_in[3];
    float*       out    = (float*)d_out;

    const int ND = in_sizes[0];          // N * DIMS
    const int B  = in_sizes[1] - 1;
    const int N  = ND / DIMS;

    float* gsum  = (float*)d_ws;         // [B][DIMS]
    float* gsqs  = gsum  + (size_t)B * DIMS;
    float* scale = gsqs  + (size_t)B * DIMS;
    float* shift = scale + (size_t)B * DIMS;

    // zero the atomic accumulators (async memset is graph-capturable)
    hipMemsetAsync(d_ws, 0, sizeof(float) * 2 * (size_t)B * DIMS, stream);

    // Pass 1: grid = B x ceil(N/CHUNK); chunks past a segment's end early-out,
    // so this is correct for arbitrary (even uneven) segment layouts.
    const int blocksPerSeg = (N + CHUNK - 1) / CHUNK;
    pin_reduce<<<dim3(B * blocksPerSeg), dim3(T1), 0, stream>>>(
        x, seqlen, gsum, gsqs, blocksPerSeg);

    // Pass 2: finalize stats
    const int stats = B * DIMS;
    pin_finalize<<<dim3((stats + 255) / 256), dim3(256), 0, stream>>>(
        seqlen, gsum, gsqs, weight, bias, scale, shift, B);

    // Pass 3: normalize stream
    const int nvec = ND / 4;
    pin_normalize<<<dim3((nvec + 255) / 256), dim3(256), 0, stream>>>(
        x, seqlen, scale, shift, out, B, nvec);
}